// GCNClassifier_25701084299499
// MI455X (gfx1250) — compile-verified
//
#include <hip/hip_runtime.h>
#include <hip/hip_bf16.h>

typedef __attribute__((ext_vector_type(2))) float v2f;
typedef __attribute__((ext_vector_type(8))) float v8f;

#define HIDDEN 64
#define N_CLS 3
#define N_GRAPHS_C 1024

// ---------------------------------------------------------------- utilities
__global__ void fill_f32(float* __restrict__ p, long n, float v) {
  long i = (long)blockIdx.x * blockDim.x + threadIdx.x;
  long stride = (long)gridDim.x * blockDim.x;
  for (; i < n; i += stride) p[i] = v;
}

__global__ void deg_count(const int* __restrict__ dst, float* __restrict__ deg, int n_edges) {
  int e = blockIdx.x * blockDim.x + threadIdx.x;
  if (e < n_edges) atomicAdd(&deg[dst[e]], 1.0f);
}

__global__ void deg_finalize(float* __restrict__ deg_dinv, int n) {
  int i = blockIdx.x * blockDim.x + threadIdx.x;
  if (i < n) deg_dinv[i] = rsqrtf(fmaxf(deg_dinv[i], 1.0f));
}

// ------------------------------------------------- f32 WMMA GEMM: [M,K]x[K,64]
// block = 128 threads = 4 waves; each wave computes a 16x16 tile of the output.
// A tile (16 rows x K) is staged through LDS in 16x64 chunks (+4 pad per row).
template <int K>
__global__ __launch_bounds__(128) void gemm_wmma(const float* __restrict__ A,
                                                 const float* __restrict__ W,
                                                 float* __restrict__ Out) {
  constexpr int KC = 64;                 // K-chunk staged in LDS
  constexpr int LDA = KC + 4;            // padded LDS row stride (floats)
  __shared__ float sA[16 * LDA];

  const int tid = threadIdx.x;
  const int wave = tid >> 5;
  const int lane = tid & 31;
  const int half = lane >> 4;            // 0: lanes 0-15, 1: lanes 16-31
  const int l15  = lane & 15;
  const int rowBase = blockIdx.x * 16;
  const int colBase = wave * 16;

  v8f acc = {};
  for (int kc = 0; kc < K; kc += KC) {
    __syncthreads();
    // stage 16 x KC chunk of A, coalesced float4 loads
    for (int idx = tid; idx < 16 * KC / 4; idx += 128) {
      int r  = idx / (KC / 4);
      int c4 = idx % (KC / 4);
      float4 v = *reinterpret_cast<const float4*>(A + (long)(rowBase + r) * K + kc + c4 * 4);
      float* dp = &sA[r * LDA + c4 * 4];
      dp[0] = v.x; dp[1] = v.y; dp[2] = v.z; dp[3] = v.w;
    }
    __syncthreads();

    #pragma unroll
    for (int k0 = 0; k0 < KC; k0 += 4) {
      const int ka = k0 + half * 2;      // lanes 16-31 hold K=+2,+3 (ISA 7.12.2)
      v2f a, b;
      const float* ap = &sA[l15 * LDA + ka];
      a.x = ap[0];
      a.y = ap[1];
      const int kg = kc + ka;
      const float* bp = &W[(long)kg * HIDDEN + colBase + l15];
      b.x = bp[0];
      b.y = bp[HIDDEN];
      acc = __builtin_amdgcn_wmma_f32_16x16x4_f32(false, a, false, b, (short)0, acc,
                                                  false, false);
    }
  }

  // D layout: VGPR r -> lanes 0-15: M=r, lanes 16-31: M=r+8; N=lane&15
  #pragma unroll
  for (int r = 0; r < 8; ++r) {
    int row = rowBase + r + half * 8;
    Out[(long)row * HIDDEN + colBase + l15] = acc[r];
  }
}

// ------------------------------------------- edge scatter: one wave per edge
__global__ __launch_bounds__(256) void scatter_edges(const float* __restrict__ h,
                                                     const int* __restrict__ src,
                                                     const int* __restrict__ dst,
                                                     const float* __restrict__ dinv,
                                                     float* __restrict__ agg,
                                                     int n_edges) {
  int e = blockIdx.x * (blockDim.x >> 5) + (threadIdx.x >> 5);
  int lane = threadIdx.x & 31;
  if (e >= n_edges) return;
  int s = src[e];
  int d = dst[e];
  float norm = dinv[s] * dinv[d];
  const float* hp = h + (long)s * HIDDEN;
  float* ap = agg + (long)d * HIDDEN;
  atomicAdd(&ap[lane],      hp[lane]      * norm);
  atomicAdd(&ap[lane + 32], hp[lane + 32] * norm);
}

// ------------- fused: self-loop term + bias + ReLU, in place on agg buffer
__global__ void bias_relu_selfloop(float* __restrict__ agg, const float* __restrict__ h,
                                   const float* __restrict__ dinv,
                                   const float* __restrict__ bias, int n_nodes) {
  long idx = (long)blockIdx.x * blockDim.x + threadIdx.x;
  if (idx >= (long)n_nodes * HIDDEN) return;
  int i = (int)(idx >> 6);
  int f = (int)(idx & 63);
  float di = dinv[i];
  float v = agg[idx] + h[idx] * di * di + bias[f];
  agg[idx] = fmaxf(v, 0.0f);
}

// ----------------------------------------------------- global mean pool (sum)
__global__ void pool_sum(const float* __restrict__ h, const int* __restrict__ batch,
                         float* __restrict__ gsum, float* __restrict__ cnt, int n_nodes) {
  long idx = (long)blockIdx.x * blockDim.x + threadIdx.x;
  if (idx >= (long)n_nodes * HIDDEN) return;
  int i = (int)(idx >> 6);
  int f = (int)(idx & 63);
  int b = batch[i];
  atomicAdd(&gsum[(long)b * HIDDEN + f], h[idx]);
  if (f == 0) atomicAdd(&cnt[b], 1.0f);
}

// ----------------------------------------------------------- final classifier
__global__ void classifier(const float* __restrict__ gsum, const float* __restrict__ cnt,
                           const float* __restrict__ Wl, const float* __restrict__ bl,
                           float* __restrict__ out, int n_graphs) {
  int idx = blockIdx.x * blockDim.x + threadIdx.x;
  if (idx >= n_graphs * N_CLS) return;
  int g = idx / N_CLS;
  int c = idx % N_CLS;
  float inv = 1.0f / fmaxf(cnt[g], 1.0f);
  float acc = bl[c];
  #pragma unroll 8
  for (int k = 0; k < HIDDEN; ++k)
    acc += gsum[(long)g * HIDDEN + k] * inv * Wl[k * N_CLS + c];
  out[idx] = acc;
}

// ---------------------------------------------------------------------------
extern "C" void kernel_launch(void* const* d_in, const int* in_sizes, int n_in,
                              void* d_out, int out_size, void* d_ws, size_t ws_size,
                              hipStream_t stream) {
  const float* x     = (const float*)d_in[0];
  const float* W1    = (const float*)d_in[1];
  const float* b1    = (const float*)d_in[2];
  const float* W2    = (const float*)d_in[3];
  const float* b2    = (const float*)d_in[4];
  const float* Wl    = (const float*)d_in[5];
  const float* bl    = (const float*)d_in[6];
  const int*   ei    = (const int*)d_in[7];
  const int*   batch = (const int*)d_in[8];

  const int n_nodes  = in_sizes[8];            // 100000 (divisible by 16)
  const int n_edges  = in_sizes[7] / 2;        // 3200000
  const int n_graphs = N_GRAPHS_C;
  const int* src = ei;
  const int* dst = ei + n_edges;

  // carve workspace (256B aligned)
  char* ws = (char*)d_ws;
  size_t off = 0;
  auto carve = [&](size_t bytes) -> void* {
    void* p = ws + off;
    off += (bytes + 255) & ~(size_t)255;
    return p;
  };
  float* bufA = (float*)carve((size_t)n_nodes * HIDDEN * sizeof(float));   // h (GEMM out)
  float* bufB = (float*)carve((size_t)n_nodes * HIDDEN * sizeof(float));   // agg / activations
  float* dinv = (float*)carve((size_t)n_nodes * sizeof(float));            // deg -> dinv
  float* gsum = (float*)carve((size_t)n_graphs * HIDDEN * sizeof(float));
  float* cnt  = (float*)carve((size_t)n_graphs * sizeof(float));

  const long nh = (long)n_nodes * HIDDEN;
  const int  nh_blocks = (int)((nh + 255) / 256);
  const int  scatter_blocks = (n_edges + 7) / 8;   // 8 waves (edges) per 256-thread block
  const int  mtiles = n_nodes / 16;                // 6250

  // degrees (self-loop = 1.0 baseline) -> dinv
  fill_f32<<<1024, 256, 0, stream>>>(dinv, n_nodes, 1.0f);
  deg_count<<<(n_edges + 255) / 256, 256, 0, stream>>>(dst, dinv, n_edges);
  deg_finalize<<<(n_nodes + 255) / 256, 256, 0, stream>>>(dinv, n_nodes);

  // layer 1: h1 = x @ W1 ; agg ; +self-loop +bias ; relu   (a1 in bufB)
  gemm_wmma<256><<<mtiles, 128, 0, stream>>>(x, W1, bufA);
  fill_f32<<<1024, 256, 0, stream>>>(bufB, nh, 0.0f);
  scatter_edges<<<scatter_blocks, 256, 0, stream>>>(bufA, src, dst, dinv, bufB, n_edges);
  bias_relu_selfloop<<<nh_blocks, 256, 0, stream>>>(bufB, bufA, dinv, b1, n_nodes);

  // layer 2: h2 = a1 @ W2 ; agg ; +self-loop +bias ; relu  (a2 in bufB)
  gemm_wmma<64><<<mtiles, 128, 0, stream>>>(bufB, W2, bufA);
  fill_f32<<<1024, 256, 0, stream>>>(bufB, nh, 0.0f);
  scatter_edges<<<scatter_blocks, 256, 0, stream>>>(bufA, src, dst, dinv, bufB, n_edges);
  bias_relu_selfloop<<<nh_blocks, 256, 0, stream>>>(bufB, bufA, dinv, b2, n_nodes);

  // global mean pool + classifier
  fill_f32<<<256, 256, 0, stream>>>(gsum, (long)n_graphs * HIDDEN, 0.0f);
  fill_f32<<<8, 256, 0, stream>>>(cnt, n_graphs, 0.0f);
  pool_sum<<<nh_blocks, 256, 0, stream>>>(bufB, batch, gsum, cnt, n_nodes);
  classifier<<<(n_graphs * N_CLS + 127) / 128, 128, 0, stream>>>(gsum, cnt, Wl, bl,
                                                                 (float*)d_out, n_graphs);
}